// EtoC_58016418234801
// MI455X (gfx1250) — compile-verified
//
#include <hip/hip_runtime.h>
#include <math.h>

typedef __attribute__((ext_vector_type(2))) float v2f;
typedef __attribute__((ext_vector_type(4))) float v4f;
typedef __attribute__((ext_vector_type(8))) float v8f;

#define DY   64
#define DC   256
#define NROW 8192          // bs * n * n = 8*32*32
#define WPAD 66            // LDS row stride for W (pad to avoid bank conflicts)

// --------------------------------------------------------------------------
// Kernel 1: EW[r, c] = sum_d X[r, d] * W[c, d]   (8192x64 @ 64x256 GEMM)
// One wave per 16-row tile; A (16x4 f32 fragments, K=64 -> 16 chunks) kept in
// registers; B streamed from LDS; 16 column tiles per wave.
// --------------------------------------------------------------------------
__global__ __launch_bounds__(256) void ew_gemm_kernel(const float* __restrict__ X,
                                                      const float* __restrict__ W,
                                                      float* __restrict__ EW) {
    __shared__ float lds_w[DC * WPAD];
    const int tid = threadIdx.x;

    // Stage W (256x64 f32 = 64 KB) into LDS as float2, padded rows.
    const v2f* W2 = (const v2f*)W;
    for (int idx = tid; idx < DC * DY / 2; idx += 256) {
        int c = idx >> 5;            // 32 float2 per row of 64
        int d = (idx & 31) * 2;
        *(v2f*)&lds_w[c * WPAD + d] = W2[idx];
    }
    __syncthreads();

    const int wave = tid >> 5;
    const int lane = tid & 31;
    const int lrow = lane & 15;      // M (or N) index within tile
    const int hi   = lane >> 4;      // selects K pair {0,1} vs {2,3}

    const int r0 = blockIdx.x * 128 + wave * 16;

    // A fragments: chunk kk covers K = 4kk..4kk+3
    v2f a[16];
    const float* xrow = X + (size_t)(r0 + lrow) * DY;
    #pragma unroll
    for (int kk = 0; kk < 16; ++kk)
        a[kk] = *(const v2f*)&xrow[kk * 4 + 2 * hi];

    for (int ct = 0; ct < 16; ++ct) {
        const int c = ct * 16 + lrow;
        const float* wrow = &lds_w[c * WPAD + 2 * hi];
        v8f acc = {};
        #pragma unroll
        for (int kk = 0; kk < 16; ++kk) {
            v2f bfrag = *(const v2f*)&wrow[kk * 4];
            acc = __builtin_amdgcn_wmma_f32_16x16x4_f32(
                false, a[kk], false, bfrag, (short)0, acc, false, false);
        }
        // D layout: VGPR v -> row (v + 8*hi), col = lane&15 within tile
        const int rowbase = r0 + 8 * hi;
        #pragma unroll
        for (int v = 0; v < 8; ++v)
            EW[(size_t)(rowbase + v) * DC + c] = acc[v];
    }
}

// --------------------------------------------------------------------------
// Kernel 2: PW[pos, c] = sum_d pe[pos, d] * W[c, d]   (64x64 @ 64x256)
// pe generated on the fly (sinusoidal PE, d_model=64, max_len=64).
// --------------------------------------------------------------------------
__device__ __forceinline__ float pe_val(int pos, int d) {
    // div_term[k] = exp(-(2k) * ln(10000)/64); col 2k = sin, col 2k+1 = cos
    const float dt = __expf(-(float)(d & ~1) * 0.14391156831212787f);
    const float ang = (float)pos * dt;
    return (d & 1) ? __cosf(ang) : __sinf(ang);
}

__global__ __launch_bounds__(128) void pw_gemm_kernel(const float* __restrict__ W,
                                                      float* __restrict__ PW) {
    const int tid  = threadIdx.x;
    const int wave = tid >> 5;
    const int lane = tid & 31;
    const int lrow = lane & 15;
    const int hi   = lane >> 4;
    const int r0   = wave * 16;          // pos tile base (4 waves cover 64 rows)
    const int pos  = r0 + lrow;

    v2f a[16];
    #pragma unroll
    for (int kk = 0; kk < 16; ++kk) {
        const int d0 = kk * 4 + 2 * hi;
        a[kk].x = pe_val(pos, d0);
        a[kk].y = pe_val(pos, d0 + 1);
    }

    for (int ct = 0; ct < 16; ++ct) {
        const int c = ct * 16 + lrow;
        v8f acc = {};
        #pragma unroll
        for (int kk = 0; kk < 16; ++kk) {
            v2f bfrag = *(const v2f*)&W[c * DY + kk * 4 + 2 * hi];
            acc = __builtin_amdgcn_wmma_f32_16x16x4_f32(
                false, a[kk], false, bfrag, (short)0, acc, false, false);
        }
        const int rowbase = r0 + 8 * hi;
        #pragma unroll
        for (int v = 0; v < 8; ++v)
            PW[(rowbase + v) * DC + c] = acc[v];
    }
}

// --------------------------------------------------------------------------
// Kernel 3: streaming broadcast + mask, the HBM-bound part (512 MB out).
// out[b, i', j', c] = m1[b, i'/8] * m2[b, j'%32] * (EW[b,i,j,c] + PW[p,q,c])
// Block = (j, i', b); 256 threads: 64 float4 columns x 4 q-groups.
// Non-temporal b128 stores: output never reread, keep L2 for EW.
// --------------------------------------------------------------------------
__global__ __launch_bounds__(256) void expand_kernel(const float* __restrict__ EW,
                                                     const float* __restrict__ PW,
                                                     const float* __restrict__ m1,
                                                     const float* __restrict__ m2,
                                                     float* __restrict__ out) {
    const int j   = blockIdx.x;      // 0..31
    const int ip  = blockIdx.y;      // i' 0..255
    const int b   = blockIdx.z;      // 0..7
    const int tid = threadIdx.x;
    const int cv  = tid & 63;        // float4 column index (c = 4*cv)
    const int qg  = tid >> 6;        // 0..3

    const int i = ip >> 3;
    const int p = ip & 7;
    const float s1 = m1[b * 32 + i];

    const v4f* EW4 = (const v4f*)EW;
    const v4f* PW4 = (const v4f*)PW;
    const v4f ew = EW4[(size_t)((b * 32 + i) * 32 + j) * 64 + cv];

    #pragma unroll
    for (int qq = 0; qq < 2; ++qq) {
        const int q  = qg * 2 + qq;            // 0..7 across the block
        const int jp = j * 8 + q;
        const float s = s1 * m2[b * 32 + (jp & 31)];   // mask2 uses j' % n !
        const v4f pw = PW4[(p * 8 + q) * 64 + cv];
        v4f val;
        val.x = s * (ew.x + pw.x);
        val.y = s * (ew.y + pw.y);
        val.z = s * (ew.z + pw.z);
        val.w = s * (ew.w + pw.w);
        const size_t o = ((size_t)((b * 256 + ip) * 256 + jp)) * 64 + cv;
        __builtin_nontemporal_store(val, (v4f*)out + o);
    }
}

extern "C" void kernel_launch(void* const* d_in, const int* in_sizes, int n_in,
                              void* d_out, int out_size, void* d_ws, size_t ws_size,
                              hipStream_t stream) {
    const float* E  = (const float*)d_in[0];   // (8,32,32,64)
    const float* m1 = (const float*)d_in[1];   // (8,32,1,1) -> 256
    const float* m2 = (const float*)d_in[2];   // (8,1,32,1) -> 256
    const float* W  = (const float*)d_in[3];   // (256,64)
    float* out = (float*)d_out;                // (8,256,256,256)

    float* EW = (float*)d_ws;                  // 8192*256 f32 = 8 MB
    float* PW = EW + (size_t)NROW * DC;        // 64*256 f32 = 64 KB

    ew_gemm_kernel<<<NROW / 128, 256, 0, stream>>>(E, W, EW);
    pw_gemm_kernel<<<1, 128, 0, stream>>>(W, PW);

    dim3 g2(32, 256, 8);
    expand_kernel<<<g2, 256, 0, stream>>>(EW, PW, m1, m2, out);
}